// MatchingNet_81810537055144
// MI455X (gfx1250) — compile-verified
//
#include <hip/hip_runtime.h>
#include <hip/hip_bf16.h>

// MatchingNet fused:  out[M,C] = diag(1/||fX||) * fX @ W^T
//   W[c,:] = sum_{n: tgt[n]==c} gS[n,:] / ||gS[n]||
// Identical (up to fp reassociation) to sim @ onehot in the reference.
// MI455X roofline: ~82 MB HBM traffic -> ~3.5us @ 23.3 TB/s; compute only
// ~1.07 GFLOP, so full f32 precision via V_WMMA_F32_16X16X4_F32.
// B (W, 256 KB) is staged block-wide into LDS with GLOBAL_LOAD_ASYNC_TO_LDS
// (ASYNCcnt) so the 4 waves of a block share one copy instead of 4x L2 reads.

typedef __attribute__((ext_vector_type(2))) float v2f;
typedef __attribute__((ext_vector_type(8))) float v8f;

static constexpr int D_DIM   = 1024;             // feature dim
static constexpr int C_CLS   = 64;               // classes
static constexpr int KSPLIT  = 4;                // K-parallelism in the GEMM
static constexpr int KCHUNK  = D_DIM / KSPLIT;   // 256 K per block
static constexpr int KSUB    = 128;              // K per LDS staging phase
static constexpr int BSTRIDE = KSUB + 2;         // LDS row pad -> <=2-way bank conflicts

// ---------------------------------------------------------------- zero ----
__global__ void mn_zero_kernel(float* __restrict__ p, int n) {
  int i = blockIdx.x * blockDim.x + threadIdx.x;
  if (i < n) p[i] = 0.0f;
}

// ------------------------------------------------- reciprocal row norms ----
// One wave32 per row; rows [0,N) -> gS/rg, rows [N,N+M) -> fX/rf.
__global__ void mn_rnorm_kernel(const float* __restrict__ gS,
                                const float* __restrict__ fX,
                                float* __restrict__ rg,
                                float* __restrict__ rf,
                                int N, int M) {
  const int wave = (int)((blockIdx.x * blockDim.x + threadIdx.x) >> 5);
  const int lane = threadIdx.x & 31;
  if (wave >= N + M) return;
  const float* __restrict__ row =
      (wave < N) ? (gS + (size_t)wave * D_DIM)
                 : (fX + (size_t)(wave - N) * D_DIM);
  float s = 0.0f;
#pragma unroll
  for (int i = 0; i < D_DIM / 128; ++i) {        // 8 x float4 per lane
    float4 v = ((const float4*)row)[lane + 32 * i];
    s = fmaf(v.x, v.x, s); s = fmaf(v.y, v.y, s);
    s = fmaf(v.z, v.z, s); s = fmaf(v.w, v.w, s);
  }
#pragma unroll
  for (int off = 16; off > 0; off >>= 1) s += __shfl_xor(s, off, 32);
  if (lane == 0) {
    float r = rsqrtf(fmaxf(s, 1e-16f));          // eps clamp never fires for D=1024 gaussians
    if (wave < N) rg[wave] = r; else rf[wave - N] = r;
  }
}

// ---------------------------------------------- class-aggregated W build ----
// One block per support row: W[tgt[n], d] += gS[n,d] * rg[n].
__global__ void mn_build_w_kernel(const float* __restrict__ gS,
                                  const int*   __restrict__ tgt,
                                  const float* __restrict__ rg,
                                  float* __restrict__ W) {
  const int n = blockIdx.x;
  const int c = tgt[n];
  const float rinv = rg[n];
  const float* __restrict__ row  = gS + (size_t)n * D_DIM;
  float* __restrict__       wrow = W + (size_t)c * D_DIM;
  for (int d = threadIdx.x; d < D_DIM; d += blockDim.x)
    atomicAdd(&wrow[d], row[d] * rinv);
}

// --------------------------------------------------------- WMMA GEMM ------
// out[M,64] += rf[m] * (fX[M,D] @ W[64,D]^T), f32 WMMA 16x16x4.
// Block = 4 waves sharing one K-segment, covering 4 adjacent 16-row strips.
// The 64x128 B-slab is staged into LDS once per phase via async-to-LDS DMA.
__global__ __launch_bounds__(128)
void mn_gemm_kernel(const float* __restrict__ fX,
                    const float* __restrict__ W,
                    const float* __restrict__ rf,
                    float* __restrict__ out, int M) {
  __shared__ float Bs[C_CLS * BSTRIDE];          // 64 x (128+2) f32 = 33 KB

  const int tid   = threadIdx.x;
  const int lane  = tid & 31;
  const int widx  = tid >> 5;
  const int kseg  = blockIdx.x & (KSPLIT - 1);   // all 4 waves share kseg
  const int sg    = blockIdx.x >> 2;             // strip group (4 strips)
  const int m0    = (sg * 4 + widx) * 16;
  const int lo = lane & 15, hi = lane >> 4;
  const int ks = kseg * KCHUNK;

  // ISA fragment layouts (CDNA5 ISA):
  //  A 16x4 f32: lane(lo,hi) holds A[M=lo, K=2*hi+{0,1}]  -> one b64 load
  //  B  4x16   : lane(lo,hi) holds B[K=2*hi+{0,1}, N=lo] = W[c0+lo, k+2*hi+{0,1}]
  const float* __restrict__ aptr = fX + (size_t)(m0 + lo) * D_DIM + ks + 2 * hi;
  const float* __restrict__ Wk   = W + ks;                 // uniform SGPR base
  // Low 32 bits of a generic pointer into LDS == wave-relative LDS byte addr.
  const unsigned lds0 = (unsigned)(uintptr_t)&Bs[0];

  v8f acc0 = {}, acc1 = {}, acc2 = {}, acc3 = {};

  for (int ph = 0; ph < KCHUNK / KSUB; ++ph) {
    // ---- stage B slab: 64 classes x 128 K-cols (32 KB) via async DMA ----
    // 2048 float4 elements, 16 per thread, fully coalesced b128 transfers.
#pragma unroll
    for (int i = 0; i < (C_CLS * KSUB / 4) / 128; ++i) {
      const int e = tid + 128 * i;
      const int c = e >> 5;                      // 32 float4 per class row
      const int q = e & 31;
      const unsigned voff = (unsigned)(c * D_DIM + ph * KSUB + q * 4) * 4u;
      const unsigned ldsb = lds0 + (unsigned)(c * BSTRIDE + q * 4) * 4u;
      asm volatile("global_load_async_to_lds_b128 %0, %1, %2"
                   :: "v"(ldsb), "v"(voff), "s"(Wk) : "memory");
    }
    asm volatile("s_wait_asynccnt 0x0" ::: "memory");  // my DMAs landed
    __syncthreads();                                    // everyone's landed

    // ---- compute 32 WMMA K-steps against the staged slab ----
#pragma unroll 4
    for (int kk = 0; kk < KSUB; kk += 4) {
      __builtin_prefetch(aptr + ph * KSUB + kk + 64, 0, 1);  // global_prefetch_b8
      v2f a  = *(const v2f*)(aptr + ph * KSUB + kk);
      v2f b0 = *(const v2f*)&Bs[(0 * 16 + lo) * BSTRIDE + kk + 2 * hi];
      v2f b1 = *(const v2f*)&Bs[(1 * 16 + lo) * BSTRIDE + kk + 2 * hi];
      v2f b2 = *(const v2f*)&Bs[(2 * 16 + lo) * BSTRIDE + kk + 2 * hi];
      v2f b3 = *(const v2f*)&Bs[(3 * 16 + lo) * BSTRIDE + kk + 2 * hi];
      acc0 = __builtin_amdgcn_wmma_f32_16x16x4_f32(false, a, false, b0, (short)0, acc0, false, false);
      acc1 = __builtin_amdgcn_wmma_f32_16x16x4_f32(false, a, false, b1, (short)0, acc1, false, false);
      acc2 = __builtin_amdgcn_wmma_f32_16x16x4_f32(false, a, false, b2, (short)0, acc2, false, false);
      acc3 = __builtin_amdgcn_wmma_f32_16x16x4_f32(false, a, false, b3, (short)0, acc3, false, false);
    }
    __syncthreads();                             // before next phase overwrites Bs
  }

  // C/D layout: VGPR r, lane(lo,hi) -> row m0 + r + 8*hi, col ct*16 + lo.
#pragma unroll
  for (int r = 0; r < 8; ++r) {
    const int m = m0 + r + 8 * hi;
    const float s = rf[m];
    float* __restrict__ orow = out + (size_t)m * C_CLS + lo;
    atomicAdd(orow +  0, acc0[r] * s);
    atomicAdd(orow + 16, acc1[r] * s);
    atomicAdd(orow + 32, acc2[r] * s);
    atomicAdd(orow + 48, acc3[r] * s);
  }
}

// ------------------------------------------------------------- launch -----
extern "C" void kernel_launch(void* const* d_in, const int* in_sizes, int n_in,
                              void* d_out, int out_size, void* d_ws, size_t ws_size,
                              hipStream_t stream) {
  const float* gS  = (const float*)d_in[0];   // [N, 1024] f32
  const float* fX  = (const float*)d_in[1];   // [M, 1024] f32
  const int*   tgt = (const int*)d_in[2];     // [N] class ids
  // d_in[3] = nClasses (device scalar) -- fixed at 64 per reference.

  const int N = in_sizes[0] / D_DIM;          // 4096
  const int M = in_sizes[1] / D_DIM;          // 8192

  // Workspace: W[64*1024] | rg[N] | rf[M]  (~304 KB)
  float* W  = (float*)d_ws;
  float* rg = W + (size_t)C_CLS * D_DIM;
  float* rf = rg + N;
  float* out = (float*)d_out;

  const int wElems = C_CLS * D_DIM;
  mn_zero_kernel<<<(wElems + 255) / 256, 256, 0, stream>>>(W, wElems);
  mn_zero_kernel<<<(out_size + 255) / 256, 256, 0, stream>>>(out, out_size);

  const int rows = N + M;
  mn_rnorm_kernel<<<(rows * 32 + 255) / 256, 256, 0, stream>>>(gS, fX, rg, rf, N, M);

  mn_build_w_kernel<<<N, 256, 0, stream>>>(gS, tgt, rg, W);

  // (M/64 strip-groups) * KSPLIT K-segments, 4 waves (64 rows) per block.
  mn_gemm_kernel<<<(M / 64) * KSPLIT, 128, 0, stream>>>(fX, W, rf, out, M);
}